// SagPooling_78520592105781
// MI455X (gfx1250) — compile-verified
//
#include <hip/hip_runtime.h>

// SagPooling (softmax pool): out[seg] = segsum(Y * exp(Yatt)) / segsum(exp(Yatt))
// T=1e6 rows, D=128 features, N=16384 segments, e_map sorted.
// Pure HBM-bandwidth problem (~528 MB -> ~23us roofline). Weighted segment
// reduction mapped onto V_WMMA_F32_16X16X4_F32: A = attention weights
// broadcast over M, B = 16-feature tile of Y for 4 consecutive rows.
// Software pipeline, unrolled x2 with ping-pong B buffers (no rotation moves):
// B data loaded 1 chunk ahead, attention values 2 stages ahead, scalar
// (readfirstlane) loop control so EXEC provably stays all-ones for WMMA.

typedef __attribute__((ext_vector_type(2))) float v2f;
typedef __attribute__((ext_vector_type(8))) float v8f;

#define FEAT 128
#define WAVES_PER_BLOCK 8

__device__ __forceinline__ int lower_bound_i32(const int* __restrict__ a, int n, int key) {
    int lo = 0, hi = n;
    while (lo < hi) {
        int mid = (lo + hi) >> 1;
        if (a[mid] < key) lo = mid + 1; else hi = mid;
    }
    return lo;
}

__global__ __launch_bounds__(WAVES_PER_BLOCK * 32)
void sag_pool_wmma_kernel(const float* __restrict__ Y,
                          const float* __restrict__ Yatt,
                          const int*   __restrict__ emap,
                          float*       __restrict__ out,
                          int T, int N) {
    const int lane = threadIdx.x & 31;
    const int wave = threadIdx.x >> 5;
    const int seg  = blockIdx.x * WAVES_PER_BLOCK + wave;
    if (seg >= N) return;  // uniform per wave

    // Segment boundaries (sorted e_map); force wave-uniform -> scalar loop control.
    const int start = __builtin_amdgcn_readfirstlane(lower_bound_i32(emap, T, seg));
    const int end   = __builtin_amdgcn_readfirstlane(lower_bound_i32(emap, T, seg + 1));

    const int half = lane >> 4;   // 0 -> K rows {0,1}; 1 -> K rows {2,3}
    const int n    = lane & 15;   // N column (feature within tile)
    const int tmax = T - 1;

    v8f acc[8];
#pragma unroll
    for (int j = 0; j < 8; ++j) acc[j] = (v8f){0.f,0.f,0.f,0.f,0.f,0.f,0.f,0.f};
    v8f accw = (v8f){0.f,0.f,0.f,0.f,0.f,0.f,0.f,0.f};
    const v2f bones = {1.0f, 1.0f};

    // All chunk loads are index-clamped -> safe even past segment/array end;
    // out-of-segment rows get weight 0, so clamped data contributes nothing.
    auto load_att = [&](int t, float& aa, float& ab) {
        int r0 = min(t + 2 * half, tmax);
        int r1 = min(t + 2 * half + 1, tmax);
        aa = Yatt[r0];
        ab = Yatt[r1];
    };
    auto load_b = [&](int t, v2f* bb) {
        int r0 = min(t + 2 * half, tmax);
        int r1 = min(t + 2 * half + 1, tmax);
        const float* pa = Y + (size_t)r0 * FEAT + n;
        const float* pb = Y + (size_t)r1 * FEAT + n;
#pragma unroll
        for (int j = 0; j < 8; ++j) { bb[j].x = pa[j * 16]; bb[j].y = pb[j * 16]; }
    };
    // Branchless: unconditional v_exp, then v_cndmask to zero rows past end.
    auto make_w = [&](int t, float aa, float ab) -> v2f {
        float ea = __expf(aa);
        float eb = __expf(ab);
        int r0 = t + 2 * half;
        int r1 = r0 + 1;
        v2f a;
        a.x = (r0 < end) ? ea : 0.0f;
        a.y = (r1 < end) ? eb : 0.0f;
        return a;   // A[m][k] = w[t+k], broadcast over all M rows
    };

    if (start < end) {
        v2f b0[8], b1[8];          // ping-pong B buffers (no rotation copies)
        float an0, an1;            // raw att for the next chunk to be weighted
        v2f acur;                  // weights for the chunk processed next
        {
            float p0, p1;
            load_att(start, p0, p1);
            load_b(start, b0);
            load_att(start + 4, an0, an1);
            acur = make_w(start, p0, p1);
        }

        for (int t0 = start; t0 < end; t0 += 8) {
            // ---- stage A: process chunk t0 (b0); prefetch chunk t0+4 ----
            float af0, af1;
            load_att(t0 + 8, af0, af1);
            load_b(t0 + 4, b1);
            v2f anxt = make_w(t0 + 4, an0, an1);   // att loaded 2 stages ago
#pragma unroll
            for (int j = 0; j < 8; ++j) {
                acc[j] = __builtin_amdgcn_wmma_f32_16x16x4_f32(
                    false, acur, false, b0[j], (short)0, acc[j], false, false);
            }
            accw = __builtin_amdgcn_wmma_f32_16x16x4_f32(
                false, acur, false, bones, (short)0, accw, false, false);

            // ---- stage B: process chunk t0+4 (b1); prefetch chunk t0+8 ----
            // (If t0+4 >= end, weights are all zero: harmless, clamped loads.)
            float ag0, ag1;
            load_att(t0 + 12, ag0, ag1);
            load_b(t0 + 8, b0);
            v2f anx2 = make_w(t0 + 8, af0, af1);
#pragma unroll
            for (int j = 0; j < 8; ++j) {
                acc[j] = __builtin_amdgcn_wmma_f32_16x16x4_f32(
                    false, anxt, false, b1[j], (short)0, acc[j], false, false);
            }
            accw = __builtin_amdgcn_wmma_f32_16x16x4_f32(
                false, anxt, false, bones, (short)0, accw, false, false);

            acur = anx2;
            an0 = ag0;
            an1 = ag1;
        }
    }

    // Accumulator VGPR0 holds (M = 0 or 8, N = lane&15); all M rows identical.
    const float wsum = accw[0];
    // lanes 0-15 write tiles 0..3, lanes 16-31 write tiles 4..7 (v_cndmask selects).
    float v0 = half ? acc[4][0] : acc[0][0];
    float v1 = half ? acc[5][0] : acc[1][0];
    float v2 = half ? acc[6][0] : acc[2][0];
    float v3 = half ? acc[7][0] : acc[3][0];

    float* o = out + (size_t)seg * FEAT + half * 64 + n;
    o[0]  = v0 / wsum;
    o[16] = v1 / wsum;
    o[32] = v2 / wsum;
    o[48] = v3 / wsum;
}

extern "C" void kernel_launch(void* const* d_in, const int* in_sizes, int n_in,
                              void* d_out, int out_size, void* d_ws, size_t ws_size,
                              hipStream_t stream) {
    // Inputs (setup_inputs order): X_in [T,D] (unused), Y [T,D], Y_att [T,1],
    // e_map [T], v_count [N] (only its length matters).
    const float* Y    = (const float*)d_in[1];
    const float* Yatt = (const float*)d_in[2];
    const int*   emap = (const int*)d_in[3];
    const int T = in_sizes[2];
    const int N = in_sizes[4];
    float* out = (float*)d_out;

    dim3 block(WAVES_PER_BLOCK * 32);
    dim3 grid((N + WAVES_PER_BLOCK - 1) / WAVES_PER_BLOCK);
    hipLaunchKernelGGL(sag_pool_wmma_kernel, grid, block, 0, stream,
                       Y, Yatt, emap, out, T, N);
}